// Attention_Module_5798205849936
// MI455X (gfx1250) — compile-verified
//
#include <hip/hip_runtime.h>

typedef __attribute__((ext_vector_type(2))) float v2f;
typedef __attribute__((ext_vector_type(4))) float v4f;
typedef __attribute__((ext_vector_type(8))) float v8f;

#define CIN   1024
#define COUT  64
#define HW    3136
#define NB    32
#define KCHUNK 64
#define LDS_STRIDE 68   // 68 floats = 272 B = 16*17 -> float4-aligned rows, conflict-free b64 reads

// ---------------------------------------------------------------------------
// K1: cams[n,k,hw] = relu(sum_c w[k,c] * x[n,c,hw]); also per-(n,k) spatial max
// grid = (ceil(HW/128), NB), block = 128 (4 waves; wave wv owns 2 16-col strips)
// ---------------------------------------------------------------------------
__global__ __launch_bounds__(128) void gemm_cam_kernel(
    const float* __restrict__ x, const float* __restrict__ w,
    float* __restrict__ cams, int* __restrict__ cmax)
{
  __shared__ float wlds[COUT * LDS_STRIDE];

  const int tid  = threadIdx.x;
  const int lane = tid & 31;
  const int wv   = tid >> 5;      // wave 0..3
  const int half = lane >> 4;     // 0|1
  const int l16  = lane & 15;
  const int n    = blockIdx.y;

  const int col0 = blockIdx.x * 128 + wv * 32 + l16;  // strip 0
  const int col1 = col0 + 16;                          // strip 1
  // clamp for loads: OOB strips recompute col 3135 (harmless duplicates for max)
  const int cc0 = col0 < HW ? col0 : HW - 1;
  const int cc1 = col1 < HW ? col1 : HW - 1;

  const float* __restrict__ xn = x + (size_t)n * CIN * HW;

  v8f acc00 = {}, acc01 = {}, acc10 = {}, acc11 = {};
  v8f acc20 = {}, acc21 = {}, acc30 = {}, acc31 = {};

  for (int kc = 0; kc < CIN; kc += KCHUNK) {
    // stage W[0..63][kc..kc+63] into LDS (8 v4f loads per thread)
    #pragma unroll
    for (int i = 0; i < 8; ++i) {
      const int q   = tid + i * 128;      // v4f index 0..1023
      const int row = q >> 4;             // 16 v4f per 64-wide row
      const int c4  = (q & 15) * 4;
      const v4f v = *(const v4f*)(w + (size_t)row * CIN + kc + c4);
      *(v4f*)(&wlds[row * LDS_STRIDE + c4]) = v;
    }
    __syncthreads();

    #pragma unroll
    for (int ks = 0; ks < KCHUNK / 4; ++ks) {
      // B fragments (4x16, K x N): lane -> { X[kg][col], X[kg+1][col] }
      const int kg = kc + 4 * ks + 2 * half;
      const size_t r0 = (size_t)kg * HW;
      const size_t r1 = (size_t)(kg + 1) * HW;
      v2f b0, b1;
      b0.x = __builtin_nontemporal_load(xn + r0 + cc0);
      b0.y = __builtin_nontemporal_load(xn + r1 + cc0);
      b1.x = __builtin_nontemporal_load(xn + r0 + cc1);
      b1.y = __builtin_nontemporal_load(xn + r1 + cc1);

      // A fragments (16x4): lane -> { W[m][ac], W[m][ac+1] }, shared by both strips
      const int ac = 4 * ks + 2 * half;
      const v2f a0 = *(const v2f*)&wlds[( 0 + l16) * LDS_STRIDE + ac];
      const v2f a1 = *(const v2f*)&wlds[(16 + l16) * LDS_STRIDE + ac];
      const v2f a2 = *(const v2f*)&wlds[(32 + l16) * LDS_STRIDE + ac];
      const v2f a3 = *(const v2f*)&wlds[(48 + l16) * LDS_STRIDE + ac];

      acc00 = __builtin_amdgcn_wmma_f32_16x16x4_f32(false, a0, false, b0, (short)0, acc00, false, false);
      acc01 = __builtin_amdgcn_wmma_f32_16x16x4_f32(false, a0, false, b1, (short)0, acc01, false, false);
      acc10 = __builtin_amdgcn_wmma_f32_16x16x4_f32(false, a1, false, b0, (short)0, acc10, false, false);
      acc11 = __builtin_amdgcn_wmma_f32_16x16x4_f32(false, a1, false, b1, (short)0, acc11, false, false);
      acc20 = __builtin_amdgcn_wmma_f32_16x16x4_f32(false, a2, false, b0, (short)0, acc20, false, false);
      acc21 = __builtin_amdgcn_wmma_f32_16x16x4_f32(false, a2, false, b1, (short)0, acc21, false, false);
      acc30 = __builtin_amdgcn_wmma_f32_16x16x4_f32(false, a3, false, b0, (short)0, acc30, false, false);
      acc31 = __builtin_amdgcn_wmma_f32_16x16x4_f32(false, a3, false, b1, (short)0, acc31, false, false);
    }
    __syncthreads();
  }

  // ReLU + store cams + per-channel spatial max (values >= 0 -> int atomicMax on bits)
  float* __restrict__ camn = cams + (size_t)n * COUT * HW;
  const v8f accs[4][2] = {{acc00, acc01}, {acc10, acc11}, {acc20, acc21}, {acc30, acc31}};
  #pragma unroll
  for (int mt = 0; mt < 4; ++mt) {
    #pragma unroll
    for (int j = 0; j < 8; ++j) {
      float v0 = accs[mt][0][j];
      float v1 = accs[mt][1][j];
      v0 = v0 > 0.0f ? v0 : 0.0f;
      v1 = v1 > 0.0f ? v1 : 0.0f;
      const int k = mt * 16 + half * 8 + j;       // C layout: VGPR j -> M = j + 8*half
      if (col0 < HW) camn[(size_t)k * HW + col0] = v0;
      if (col1 < HW) camn[(size_t)k * HW + col1] = v1;

      // max over this wave's 32 columns (duplicates from clamping are harmless)
      float m = fmaxf(v0, v1);
      m = fmaxf(m, __shfl_xor(m, 1, 16));
      m = fmaxf(m, __shfl_xor(m, 2, 16));
      m = fmaxf(m, __shfl_xor(m, 4, 16));
      m = fmaxf(m, __shfl_xor(m, 8, 16));
      if (l16 == 0)
        atomicMax(&cmax[n * COUT + k], __float_as_int(m));
    }
  }
}

// ---------------------------------------------------------------------------
// K2: mean_drop[n,hw] = mean_k( cams > 0.7*max ? 0 : cams )
// ---------------------------------------------------------------------------
__global__ __launch_bounds__(256) void drop_mean_kernel(
    const float* __restrict__ cams, const int* __restrict__ cmax,
    const float* __restrict__ gama, float* __restrict__ mdrop)
{
  const int pos = blockIdx.x * 256 + threadIdx.x;   // < NB*HW (exact)
  const int n  = pos / HW;
  const int hw = pos - n * HW;
  const float g = gama[0];
  const float* __restrict__ c  = cams + (size_t)n * COUT * HW + hw;
  const int*   __restrict__ mx = cmax + n * COUT;
  float s = 0.0f;
  #pragma unroll 8
  for (int k = 0; k < COUT; ++k) {
    const float thr = g * __int_as_float(mx[k]);
    const float v   = c[(size_t)k * HW];
    s += (v > thr) ? 0.0f : v;
  }
  mdrop[pos] = s * (1.0f / COUT);
}

// ---------------------------------------------------------------------------
// K3: out = x * mean_drop (broadcast over channels), v4f + non-temporal
// ---------------------------------------------------------------------------
__global__ __launch_bounds__(256) void scale_kernel(
    const float* __restrict__ x, const float* __restrict__ mdrop,
    float* __restrict__ out)
{
  const int i4 = blockIdx.x * 256 + threadIdx.x;    // v4f index
  const size_t e = (size_t)i4 * 4;
  const int n  = (int)(e / ((size_t)CIN * HW));
  const int hw = (int)(e % HW);                     // multiple of 4 (HW % 4 == 0)
  const v4f xv = __builtin_nontemporal_load((const v4f*)(x + e));
  const v4f mv = *(const v4f*)(mdrop + (size_t)n * HW + hw);
  const v4f o = xv * mv;
  __builtin_nontemporal_store(o, (v4f*)(out + e));
}

extern "C" void kernel_launch(void* const* d_in, const int* in_sizes, int n_in,
                              void* d_out, int out_size, void* d_ws, size_t ws_size,
                              hipStream_t stream) {
  const float* x    = (const float*)d_in[0];
  const float* w    = (const float*)d_in[1];   // (64,1024,1,1) row-major
  const float* gama = (const float*)d_in[2];
  float* out = (float*)d_out;

  // workspace layout: cams (32*64*3136 f32) | cmax (2048 i32) | mdrop (32*3136 f32)
  float* cams  = (float*)d_ws;
  const size_t cams_elems = (size_t)NB * COUT * HW;
  int*   cmax  = (int*)(cams + cams_elems);
  float* mdrop = (float*)(cmax + NB * COUT);

  (void)hipMemsetAsync(cmax, 0, NB * COUT * sizeof(int), stream);  // relu output >= 0

  gemm_cam_kernel<<<dim3((HW + 127) / 128, NB), 128, 0, stream>>>(x, w, cams, cmax);

  drop_mean_kernel<<<dim3((NB * HW) / 256), 256, 0, stream>>>(cams, cmax, gama, mdrop);

  const int n4 = (int)(((size_t)NB * CIN * HW / 4) / 256);   // 100,352 blocks
  scale_kernel<<<dim3(n4), 256, 0, stream>>>(x, mdrop, out);
}